// SemanticSimilarityMeasure_81595788689982
// MI455X (gfx1250) — compile-verified
//
#include <hip/hip_runtime.h>
#include <math.h>
#include <stdint.h>

typedef float v2f __attribute__((ext_vector_type(2)));
typedef float v8f __attribute__((ext_vector_type(8)));
typedef unsigned int u32x4 __attribute__((ext_vector_type(4)));
typedef int i32x4 __attribute__((ext_vector_type(4)));
typedef int i32x8 __attribute__((ext_vector_type(8)));

#if __has_builtin(__builtin_amdgcn_tensor_load_to_lds)
#define HAVE_TDM 1
#else
#define HAVE_TDM 0
#endif

#define B_      8
#define L_      2048
#define D_      256
#define P_TILE  32
#define S_TILE  64
#define LD      260      // LDS row stride (floats): 256 data + 4 pad (TDM pad_interval=256dw, pad_amount=4dw)
#define LLD     66
#define NEG_INF_ (-1e10f)
#define THREADS 256

// dynamic-LDS float counts
#define SM_PRED   (P_TILE * LD)        // 8320
#define SM_PREDW  (P_TILE * LD)        // 8320
#define SM_BUF    (64 * LD)            // 16640 (x2: double-buffered W-strip / src tile)
#define SM_LOGIT  (P_TILE * LLD)       // 2112
#define SM_COST   (P_TILE * LLD)       // 2112
#define SMEM_FLOATS (SM_PRED + SM_PREDW + 2*SM_BUF + SM_LOGIT + SM_COST \
                     + 32 + 64 + 32 + L_ + 256 + 3*256 + 32)

#if HAVE_TDM
// Issue one TDM 2-D tile load: rows x 256 fp32, row-major global -> LDS with
// 4-dword padding per 256-dword row (=> LD stride). Wave-scalar; EXEC ignored.
__device__ __forceinline__ void tdm_load_rows(const float* smem_base, float* lds_dst,
                                              const float* gsrc, int rows) {
    uint32_t lo = (uint32_t)((const char*)lds_dst - (const char*)smem_base); // dynamic LDS starts at 0
    uint64_t ga = (uint64_t)(uintptr_t)gsrc;
    u32x4 g0;
    g0[0] = 1u;                                                  // count=1, user descriptor
    g0[1] = lo;                                                  // lds_addr
    g0[2] = (uint32_t)ga;                                        // global_addr[31:0]
    g0[3] = (uint32_t)((ga >> 32) & 0x01FFFFFFu) | (2u << 30);   // global_addr[56:32] | type=2
    i32x8 g1;
    g1[0] = (int)((2u << 16)       // data_size = 4B
                | (1u << 20)       // pad_enable
                | (7u << 22)       // pad_interval: 256 dwords
                | (3u << 25));     // pad_amount:   4 dwords
    g1[1] = (int)(256u << 16);                 // tensor_dim0 = 256
    g1[2] = (int)((uint32_t)rows << 16);       // tensor_dim1 = rows
    g1[3] = (int)(256u << 16);                 // tile_dim0 = 256
    g1[4] = rows;                              // tile_dim1 = rows (tile_dim2 = 0)
    g1[5] = 256;                               // tensor_dim0_stride = 256
    g1[6] = 0;
    g1[7] = 0;
    i32x4 z4 = (i32x4)0;
#if defined(__clang_major__) && __clang_major__ >= 23
    i32x8 z8 = (i32x8)0;
    __builtin_amdgcn_tensor_load_to_lds(g0, g1, z4, z4, z8, 0);
#else
    __builtin_amdgcn_tensor_load_to_lds(g0, g1, z4, z4, 0);
#endif
}
#else
// fallback: cooperative vectorized staging
__device__ __forceinline__ void stage_rows(float* dst, const float* g, int rows, int t) {
    const float4* gv = (const float4*)g;
    int n4 = rows * 64;
    for (int j = t; j < n4; j += THREADS) {
        int row = j >> 6, c4 = j & 63;
        *(float4*)&dst[row * LD + c4 * 4] = gv[j];
    }
}
#endif

__global__ __launch_bounds__(THREADS)
void ssm_fused_kernel(const float* __restrict__ hx,   // src  [B,L,D]
                      const float* __restrict__ hy,   // pred [B,L,D]
                      const float* __restrict__ xmask,// src mask [B,L]
                      const float* __restrict__ ymask,// pred mask [B,L]
                      const float* __restrict__ Wm,   // [D,D]
                      float* __restrict__ partial)    // [gridDim.x]
{
    extern __shared__ float smem[];
    float* s_pred  = smem;                    // [32][LD]
    float* s_predw = s_pred  + SM_PRED;       // [32][LD]
    float* s_buf0  = s_predw + SM_PREDW;      // [64][LD]
    float* s_buf1  = s_buf0  + SM_BUF;        // [64][LD]
    float* s_logit = s_buf1  + SM_BUF;        // [32][LLD]
    float* s_cost  = s_logit + SM_LOGIT;      // [32][LLD]
    float* s_psq   = s_cost  + SM_COST;       // [32]
    float* s_ssq   = s_psq   + 32;            // [64]
    float* s_pm    = s_ssq   + 64;            // [32]
    float* s_smAll = s_pm    + 32;            // [2048] full src-mask row for this batch
    float* s_red   = s_smAll + L_;            // [256]
    float* s_m     = s_red   + 256;           // [256]
    float* s_d     = s_m     + 256;           // [256]
    float* s_n     = s_d     + 256;           // [256]
    float* s_rowf  = s_n     + 256;           // [32]

    const int t    = threadIdx.x;
    const int lane = t & 31;
    const int wv   = t >> 5;              // 8 waves
    const int nn   = lane & 15;
    const int koff = (lane >> 4) << 1;    // fp32 16x16x4 A/B layout K sub-offset
    const int hi   = lane >> 4;

    const int blk = blockIdx.x;
    const int b   = blk >> 6;
    const int p0  = (blk & 63) * P_TILE;

    const float* predB = hy + (size_t)b * L_ * D_;
    const float* srcB  = hx + (size_t)b * L_ * D_;

    // ---------- stage pred tile (TDM) + issue first W strip; stage masks ----------
#if HAVE_TDM
    if (wv == 0) {
        tdm_load_rows(smem, s_pred, predB + (size_t)p0 * D_, P_TILE);
        tdm_load_rows(smem, s_buf0, Wm, 64);               // W strip kb=0 prefetch
        __builtin_amdgcn_s_wait_tensorcnt(1);              // pred tile landed
    }
#else
    stage_rows(s_pred, predB + (size_t)p0 * D_, P_TILE, t);
#endif
    for (int i = t; i < L_; i += THREADS) s_smAll[i] = xmask[(size_t)b * L_ + i];
    if (t < P_TILE) s_pm[t] = ymask[(size_t)b * L_ + p0 + t];
    __syncthreads();

    // ---------- pred row squared norms ----------
    {
        int row = t >> 3, part = t & 7;
        const float* r = &s_pred[row * LD + part * 32];
        float s = 0.f;
        #pragma unroll
        for (int i = 0; i < 32; ++i) s += r[i] * r[i];
        s_red[t] = s;
    }
    __syncthreads();
    if (t < P_TILE) {
        float s = 0.f;
        #pragma unroll
        for (int i = 0; i < 8; ++i) s += s_red[t * 8 + i];
        s_psq[t] = s;
    }

    // ---------- phase 1: predW = pred @ W, double-buffered W strips ----------
    v8f accW[4] = {};
    for (int kb = 0; kb < 4; ++kb) {                // 4 strips of 64 K-rows
        __syncthreads();
#if HAVE_TDM
        if (wv == 0) {
            if (kb < 3) {
                tdm_load_rows(smem, (kb & 1) ? s_buf0 : s_buf1, Wm + (size_t)(kb + 1) * 64 * D_, 64);
                __builtin_amdgcn_s_wait_tensorcnt(1);   // strip kb done, kb+1 in flight
            } else {
                __builtin_amdgcn_s_wait_tensorcnt(0);
            }
        }
#else
        stage_rows((kb & 1) ? s_buf1 : s_buf0, Wm + (size_t)kb * 64 * D_, 64, t);
#endif
        __syncthreads();
        const float* s_w = (kb & 1) ? s_buf1 : s_buf0;
        const int e0a = wv * 16, e0b = (wv + 8) * 16;
        #pragma unroll 2
        for (int kk = 0; kk < 64; kk += 4) {
            int k = kb * 64 + kk;
            v2f a0 = *(const v2f*)&s_pred[(0 * 16 + nn) * LD + k + koff];
            v2f a1 = *(const v2f*)&s_pred[(1 * 16 + nn) * LD + k + koff];
            v2f bA, bB;
            bA.x = s_w[(kk + koff    ) * LD + e0a + nn];
            bA.y = s_w[(kk + koff + 1) * LD + e0a + nn];
            bB.x = s_w[(kk + koff    ) * LD + e0b + nn];
            bB.y = s_w[(kk + koff + 1) * LD + e0b + nn];
            accW[0] = __builtin_amdgcn_wmma_f32_16x16x4_f32(false, a0, false, bA, (short)0, accW[0], false, false);
            accW[1] = __builtin_amdgcn_wmma_f32_16x16x4_f32(false, a0, false, bB, (short)0, accW[1], false, false);
            accW[2] = __builtin_amdgcn_wmma_f32_16x16x4_f32(false, a1, false, bA, (short)0, accW[2], false, false);
            accW[3] = __builtin_amdgcn_wmma_f32_16x16x4_f32(false, a1, false, bB, (short)0, accW[3], false, false);
        }
        __syncthreads();
    }
    // spill predW tiles (C/D layout: vgpr r -> M = r + 8*hi, N = nn)
    #pragma unroll
    for (int tIdx = 0; tIdx < 4; ++tIdx) {
        int pi = tIdx >> 1;
        int e0 = (wv + 8 * (tIdx & 1)) * 16;
        #pragma unroll
        for (int r = 0; r < 8; ++r)
            s_predw[(pi * 16 + r + 8 * hi) * LD + e0 + nn] = accW[tIdx][r];
    }
#if HAVE_TDM
    if (wv == 0) tdm_load_rows(smem, s_buf0, srcB, 64);    // prefetch src tile 0
#endif

    // ---------- phase 2: fused GEMMs + online softmax, double-buffered src tiles ----------
    float om = -INFINITY, od = 0.f, on = 0.f;
    const int srow = t >> 3;
    const int ssub = t & 7;
    const int pi2 = wv >> 2, si2 = wv & 3;

    for (int it = 0; it < L_ / S_TILE; ++it) {             // 32 tiles
        __syncthreads();
#if HAVE_TDM
        if (wv == 0) {
            if (it < L_ / S_TILE - 1) {
                tdm_load_rows(smem, (it & 1) ? s_buf0 : s_buf1,
                              srcB + (size_t)(it + 1) * S_TILE * D_, 64);
                __builtin_amdgcn_s_wait_tensorcnt(1);      // tile it ready, it+1 streaming
            } else {
                __builtin_amdgcn_s_wait_tensorcnt(0);
            }
        }
#else
        stage_rows((it & 1) ? s_buf1 : s_buf0, srcB + (size_t)it * S_TILE * D_, 64, t);
#endif
        __syncthreads();
        const float* s_src = (it & 1) ? s_buf1 : s_buf0;
        const int s0 = it * S_TILE;

        // src row squared norms
        {
            int row = t >> 2, part = t & 3;
            const float* r = &s_src[row * LD + part * 64];
            float s = 0.f;
            #pragma unroll
            for (int i = 0; i < 64; ++i) s += r[i] * r[i];
            s_red[t] = s;
        }
        __syncthreads();
        if (t < S_TILE)
            s_ssq[t] = s_red[t * 4] + s_red[t * 4 + 1] + s_red[t * 4 + 2] + s_red[t * 4 + 3];
        __syncthreads();

        // two K=256 fp32-WMMA chains sharing the B (src) operand
        v8f accC = {}, accT = {};
        const float* aP = &s_pred [(pi2 * 16 + nn) * LD + koff];
        const float* aW = &s_predw[(pi2 * 16 + nn) * LD + koff];
        const float* bS = &s_src  [(si2 * 16 + nn) * LD + koff];
        #pragma unroll 4
        for (int k = 0; k < D_; k += 4) {
            v2f a_c = *(const v2f*)(aP + k);
            v2f a_t = *(const v2f*)(aW + k);
            v2f b_s = *(const v2f*)(bS + k);
            accC = __builtin_amdgcn_wmma_f32_16x16x4_f32(false, a_c, false, b_s, (short)0, accC, false, false);
            accT = __builtin_amdgcn_wmma_f32_16x16x4_f32(false, a_t, false, b_s, (short)0, accT, false, false);
        }
        #pragma unroll
        for (int r = 0; r < 8; ++r) {
            int pl = pi2 * 16 + r + 8 * hi;
            int sl = si2 * 16 + nn;
            float cost = sqrtf(fmaxf(s_psq[pl] + s_ssq[sl] - 2.0f * accC[r], 0.0f));
            s_cost [pl * LLD + sl] = cost;
            s_logit[pl * LLD + sl] = accT[r];
        }
        __syncthreads();

        // online-softmax partial update (faithful -1e10 masking semantics)
        {
            float pm = s_pm[srow];
            #pragma unroll
            for (int j = 0; j < 8; ++j) {
                int sl = ssub + 8 * j;
                float smv = s_smAll[s0 + sl];
                float l = (pm * smv == 0.0f) ? NEG_INF_ : s_logit[srow * LLD + sl];
                float c = s_cost[srow * LLD + sl];
                float mnew  = fmaxf(om, l);
                float scale = __expf(om - mnew);
                float e     = __expf(l - mnew);
                od = od * scale + e;
                on = on * scale + e * c;
                om = mnew;
            }
        }
    }

    // ---------- merge partials -> per-row flow -> block partial ----------
    s_m[t] = om; s_d[t] = od; s_n[t] = on;
    __syncthreads();
    if (t < P_TILE) {
        float M = -INFINITY;
        #pragma unroll
        for (int j = 0; j < 8; ++j) M = fmaxf(M, s_m[t * 8 + j]);
        float Dd = 0.f, Nn = 0.f;
        #pragma unroll
        for (int j = 0; j < 8; ++j) {
            float sc = __expf(s_m[t * 8 + j] - M);
            Dd += s_d[t * 8 + j] * sc;
            Nn += s_n[t * 8 + j] * sc;
        }
        s_rowf[t] = Nn / Dd;
    }
    __syncthreads();
    if (t == 0) {
        float s = 0.f;
        for (int i = 0; i < P_TILE; ++i) s += s_rowf[i];
        partial[blk] = s;
    }
}

// deterministic fixed-order reduction + separable batch normalization
__global__ __launch_bounds__(THREADS)
void ssm_finalize_kernel(const float* __restrict__ xmask,
                         const float* __restrict__ ymask,
                         const float* __restrict__ partial,
                         float* __restrict__ out)
{
    __shared__ float s_red[THREADS];
    int t = threadIdx.x;
    s_red[t] = partial[t] + partial[t + 256];
    __syncthreads();
    for (int st = 128; st > 0; st >>= 1) {
        if (t < st) s_red[t] += s_red[t + st];
        __syncthreads();
    }
    float S = s_red[0];
    __syncthreads();

    float inv = 0.f;
    for (int bb = 0; bb < B_; ++bb) {
        float sy = 0.f, sx = 0.f;
        #pragma unroll
        for (int i = 0; i < 8; ++i) {
            sy += ymask[(size_t)bb * L_ + t + 256 * i];
            sx += xmask[(size_t)bb * L_ + t + 256 * i];
        }
        s_red[t] = sy;
        __syncthreads();
        for (int st = 128; st > 0; st >>= 1) {
            if (t < st) s_red[t] += s_red[t + st];
            __syncthreads();
        }
        float n1 = s_red[0];
        __syncthreads();
        s_red[t] = sx;
        __syncthreads();
        for (int st = 128; st > 0; st >>= 1) {
            if (t < st) s_red[t] += s_red[t + st];
            __syncthreads();
        }
        float n2 = s_red[0];
        __syncthreads();
        inv += 1.0f / (n1 * n2);
    }
    if (t == 0)
        out[0] = S * inv / (float)(B_ * B_);   // LAMBDA_W = 1.0
}

extern "C" void kernel_launch(void* const* d_in, const int* in_sizes, int n_in,
                              void* d_out, int out_size, void* d_ws, size_t ws_size,
                              hipStream_t stream) {
    (void)in_sizes; (void)n_in; (void)out_size; (void)ws_size;
    const float* hx = (const float*)d_in[0];   // h_x  [8,2048,256]
    const float* hy = (const float*)d_in[1];   // h_y  [8,2048,256]
    const float* xm = (const float*)d_in[2];   // x_mask [8,2048]
    const float* ym = (const float*)d_in[3];   // y_mask [8,2048]
    const float* W  = (const float*)d_in[4];   // W [256,256]
    float* out      = (float*)d_out;
    float* partial  = (float*)d_ws;            // 512 floats used

    const int nblocks = B_ * (L_ / P_TILE);    // 512
    const size_t smem = SMEM_FLOATS * sizeof(float);  // ~224 KB dynamic LDS (WGP has 320 KB)
    ssm_fused_kernel<<<nblocks, THREADS, smem, stream>>>(hx, hy, xm, ym, W, partial);
    ssm_finalize_kernel<<<1, THREADS, 0, stream>>>(xm, ym, partial, out);
}